// MultiBoxLoss_24489903522364
// MI455X (gfx1250) — compile-verified
//
#include <hip/hip_runtime.h>
#include <hip/hip_bf16.h>
#include <math.h>

#define B_    32
#define P_    24564
#define T_    50
#define C_    81
#define K2CH  96     // ceil(P_/256)
#define K3CH  384    // ceil(P_/64)

typedef __attribute__((ext_vector_type(2))) float v2f;
typedef __attribute__((ext_vector_type(8))) float v8f;

__device__ __forceinline__ float smoothl1(float d) {
    float a = fabsf(d);
    return (a < 1.0f) ? 0.5f * d * d : a - 0.5f;
}

// ---------------------------------------------------------------------------
// K1: best prior per truth: argmax over P of IoU(truth, prior_pointform)
// grid = (T_, B_), block = 256
// ---------------------------------------------------------------------------
__global__ void mb_best_prior(const float* __restrict__ priors,
                              const float* __restrict__ boxes,
                              int* __restrict__ bpi)
{
    const int t = blockIdx.x, b = blockIdx.y, tid = threadIdx.x;
    const float tx0 = boxes[((size_t)b * T_ + t) * 4 + 0];
    const float ty0 = boxes[((size_t)b * T_ + t) * 4 + 1];
    const float tx1 = boxes[((size_t)b * T_ + t) * 4 + 2];
    const float ty1 = boxes[((size_t)b * T_ + t) * 4 + 3];
    const float areaA = (tx1 - tx0) * (ty1 - ty0);

    float bv = -1.0f;
    int bidx = 0x7fffffff;
    for (int p = tid; p < P_; p += 256) {
        float pcx = priors[p * 4 + 0], pcy = priors[p * 4 + 1];
        float pw  = priors[p * 4 + 2], ph  = priors[p * 4 + 3];
        float px0 = pcx - 0.5f * pw, py0 = pcy - 0.5f * ph;
        float px1 = pcx + 0.5f * pw, py1 = pcy + 0.5f * ph;
        float iw = fminf(tx1, px1) - fmaxf(tx0, px0);
        float ih = fminf(ty1, py1) - fmaxf(ty0, py0);
        iw = fmaxf(iw, 0.0f); ih = fmaxf(ih, 0.0f);
        float inter = iw * ih;
        float iou = inter / (areaA + pw * ph - inter);
        if (iou > bv) { bv = iou; bidx = p; }   // first occurrence wins (ascending p)
    }
    __shared__ float sv[256];
    __shared__ int   si[256];
    sv[tid] = bv; si[tid] = bidx;
    __syncthreads();
    for (int s = 128; s > 0; s >>= 1) {
        if (tid < s) {
            if (sv[tid + s] > sv[tid] ||
                (sv[tid + s] == sv[tid] && si[tid + s] < si[tid])) {
                sv[tid] = sv[tid + s]; si[tid] = si[tid + s];
            }
        }
        __syncthreads();
    }
    if (tid == 0) bpi[b * T_ + t] = si[0];
}

// ---------------------------------------------------------------------------
// K2: per-prior match + encode + smooth-L1 partials
// grid = (K2CH, B_), block = 256
// ---------------------------------------------------------------------------
__global__ void mb_match(const float* __restrict__ loc,
                         const float* __restrict__ priors,
                         const float* __restrict__ boxes,
                         const int*   __restrict__ labels,
                         const int*   __restrict__ bpi,
                         int*   __restrict__ conf_t,
                         float* __restrict__ lossl_part,
                         int*   __restrict__ poscnt_part)
{
    const int b = blockIdx.y, tid = threadIdx.x;
    const int p = blockIdx.x * 256 + tid;
    __shared__ float s_tr[T_ * 4];
    __shared__ int   s_lab[T_];
    __shared__ int   s_bpi[T_];
    if (tid < T_)     { s_lab[tid] = labels[b * T_ + tid]; s_bpi[tid] = bpi[b * T_ + tid]; }
    if (tid < T_ * 4) { s_tr[tid]  = boxes[(size_t)b * T_ * 4 + tid]; }
    __syncthreads();

    float sl1 = 0.0f;
    int   cnt = 0;
    if (p < P_) {
        float pcx = priors[p * 4 + 0], pcy = priors[p * 4 + 1];
        float pw  = priors[p * 4 + 2], ph  = priors[p * 4 + 3];
        float px0 = pcx - 0.5f * pw, py0 = pcy - 0.5f * ph;
        float px1 = pcx + 0.5f * pw, py1 = pcy + 0.5f * ph;
        float areaB = pw * ph;

        float bo = -1.0f; int bi = 0; int ft = -1;
        #pragma unroll 5
        for (int t = 0; t < T_; ++t) {
            if (s_bpi[t] == p) ft = t;                      // last t wins (scatter order)
            float tx0 = s_tr[t * 4 + 0], ty0 = s_tr[t * 4 + 1];
            float tx1 = s_tr[t * 4 + 2], ty1 = s_tr[t * 4 + 3];
            float iw = fminf(tx1, px1) - fmaxf(tx0, px0);
            float ih = fminf(ty1, py1) - fmaxf(ty0, py0);
            iw = fmaxf(iw, 0.0f); ih = fmaxf(ih, 0.0f);
            float inter = iw * ih;
            float areaA = (tx1 - tx0) * (ty1 - ty0);
            float iou = inter / (areaA + areaB - inter);
            if (iou > bo) { bo = iou; bi = t; }             // first max wins
        }
        if (ft >= 0) { bi = ft; bo = 2.0f; }                // forced bipartite match
        int conf = s_lab[bi] + 1;
        if (bo < 0.5f) conf = 0;

        float mx0 = s_tr[bi * 4 + 0], my0 = s_tr[bi * 4 + 1];
        float mx1 = s_tr[bi * 4 + 2], my1 = s_tr[bi * 4 + 3];
        float gx = ((mx0 + mx1) * 0.5f - pcx) / (0.1f * pw);
        float gy = ((my0 + my1) * 0.5f - pcy) / (0.1f * ph);
        float gw = logf((mx1 - mx0) / pw) / 0.2f;
        float gh = logf((my1 - my0) / ph) / 0.2f;

        const float* L = loc + ((size_t)b * P_ + p) * 4;
        float s = smoothl1(L[0] - gx) + smoothl1(L[1] - gy)
                + smoothl1(L[2] - gw) + smoothl1(L[3] - gh);
        if (conf > 0) { sl1 = s; cnt = 1; }
        conf_t[(size_t)b * P_ + p] = conf;
    }

    __shared__ float sf[256];
    __shared__ int   si[256];
    sf[tid] = sl1; si[tid] = cnt;
    __syncthreads();
    for (int s = 128; s > 0; s >>= 1) {
        if (tid < s) { sf[tid] += sf[tid + s]; si[tid] += si[tid + s]; }
        __syncthreads();
    }
    if (tid == 0) {
        lossl_part[b * K2CH + blockIdx.x]  = sf[0];
        poscnt_part[b * K2CH + blockIdx.x] = si[0];
    }
}

// ---------------------------------------------------------------------------
// K3: fused log-softmax cross-entropy. One wave = 16 prior rows.
// exp-sum over 81 classes via chained V_WMMA_F32_16X16X4_F32 with B = ones
// (f32 WMMA: exact precision, any column of D = row sum).
// grid = (K3CH, B_), block = 128 (4 waves -> 64 priors per block)
// ---------------------------------------------------------------------------
__global__ void mb_ce(const float* __restrict__ conf,
                      const int*   __restrict__ conf_t,
                      float* __restrict__ mine,
                      float* __restrict__ posce_part)
{
    const int b    = blockIdx.y;
    const int tid  = threadIdx.x;
    const int wave = tid >> 5;
    const int lane = tid & 31;
    const int r    = lane & 15;       // row within 16x16 tile (ISA A/C layout)
    const int hi   = lane >> 4;       // half: low lanes K={0,1}, high lanes K={2,3}
    const int p    = blockIdx.x * 64 + wave * 16 + r;
    const bool rowValid = (p < P_);
    const float* row = conf + ((size_t)b * P_ + (rowValid ? p : 0)) * C_;

    __builtin_prefetch(row, 0, 1);    // global_prefetch into cache hierarchy

    // pass 1: load this lane's classes (c = 4j + 2*hi, 4j + 2*hi + 1), row max
    float x0[21], x1[21];
    float mymax = -3.0e38f;
    #pragma unroll
    for (int j = 0; j < 21; ++j) {
        int c = 4 * j + 2 * hi;
        float a0 = (rowValid && (c     < C_)) ? row[c]     : -3.0e38f;
        float a1 = (rowValid && (c + 1 < C_)) ? row[c + 1] : -3.0e38f;
        x0[j] = a0; x1[j] = a1;
        mymax = fmaxf(mymax, fmaxf(a0, a1));
    }
    float m = fmaxf(mymax, __shfl_xor(mymax, 16));   // combine K-halves (wave32)

    // pass 2: exp + WMMA ones-reduction. A = 16x4 f32 tile per ISA layout,
    // B = all ones (4x16) -> every D column is the running row sum.
    v8f acc = {};
    v2f bones; bones.x = 1.0f; bones.y = 1.0f;
    #pragma unroll
    for (int j = 0; j < 21; ++j) {
        int c = 4 * j + 2 * hi;
        float e0 = (rowValid && (c     < C_)) ? __expf(x0[j] - m) : 0.0f;
        float e1 = (rowValid && (c + 1 < C_)) ? __expf(x1[j] - m) : 0.0f;
        v2f a; a.x = e0; a.y = e1;
        acc = __builtin_amdgcn_wmma_f32_16x16x4_f32(
            false, a, false, bones, (short)0, acc, false, false);
    }

    // extract row sum: row q(=r&7) lives in VGPR q; rows 0-7 in lanes 0-15,
    // rows 8-15 in lanes 16-31 (C/D layout). Lane pair (L, L+16) shares r.
    int q = r & 7;
    float sel = (q == 0) ? acc[0] : (q == 1) ? acc[1] : (q == 2) ? acc[2]
              : (q == 3) ? acc[3] : (q == 4) ? acc[4] : (q == 5) ? acc[5]
              : (q == 6) ? acc[6] : acc[7];
    float other = __shfl_xor(sel, 16);
    float S = (((r < 8) ? 1 : 0) == ((hi == 0) ? 1 : 0)) ? sel : other;

    float ce = 0.0f;
    bool  pos = false;
    if (hi == 0 && rowValid) {
        int tgt = conf_t[(size_t)b * P_ + p];
        float xt = row[tgt];
        ce  = (m + logf(S)) - xt;                 // logsumexp - x[target], >= 0
        pos = (tgt > 0);
        mine[(size_t)b * P_ + p] = pos ? 0.0f : ce;
    }

    __shared__ float sred[128];
    sred[tid] = (hi == 0 && rowValid && pos) ? ce : 0.0f;
    __syncthreads();
    for (int s = 64; s > 0; s >>= 1) {
        if (tid < s) sred[tid] += sred[tid + s];
        __syncthreads();
    }
    if (tid == 0) posce_part[b * K3CH + blockIdx.x] = sred[0];
}

// ---------------------------------------------------------------------------
// K4: per-batch k-th-largest bit-radix select on `mine` (nonneg floats:
// uint bit order == float order), then tie-exact top-k sum.
// grid = B_, block = 256. mine rows are L2-resident (3.1 MB total).
// ---------------------------------------------------------------------------
__global__ void mb_select(const float* __restrict__ mine,
                          const int*   __restrict__ poscnt_part,
                          float* __restrict__ negsum,
                          int*   __restrict__ numpos_ws)
{
    const int b = blockIdx.x, tid = threadIdx.x;
    __shared__ int   si[256];
    __shared__ float sf[256];
    __shared__ int   s_total;

    si[tid] = (tid < K2CH) ? poscnt_part[b * K2CH + tid] : 0;
    __syncthreads();
    for (int s = 128; s > 0; s >>= 1) {
        if (tid < s) si[tid] += si[tid + s];
        __syncthreads();
    }
    int np = si[0];
    __syncthreads();
    if (tid == 0) numpos_ws[b] = np;

    int k = 3 * np;
    if (k > P_ - 1) k = P_ - 1;
    if (k <= 0) { if (tid == 0) negsum[b] = 0.0f; return; }

    const float* rowp = mine + (size_t)b * P_;
    unsigned prefix = 0;
    int rem = k;
    for (int bit = 31; bit >= 0; --bit) {
        unsigned want = (prefix >> bit) | 1u;
        int c = 0;
        for (int i = tid; i < P_; i += 256) {
            unsigned v = __float_as_uint(rowp[i]);
            c += ((v >> bit) == want) ? 1 : 0;
        }
        si[tid] = c;
        __syncthreads();
        for (int s = 128; s > 0; s >>= 1) {
            if (tid < s) si[tid] += si[tid + s];
            __syncthreads();
        }
        if (tid == 0) s_total = si[0];
        __syncthreads();
        int total = s_total;
        if (total >= rem) prefix |= (1u << bit);
        else              rem -= total;
        __syncthreads();
    }
    // prefix = bit pattern of the k-th largest value (threshold)
    int   cg = 0;
    float sg = 0.0f;
    for (int i = tid; i < P_; i += 256) {
        float fv = rowp[i];
        if (__float_as_uint(fv) > prefix) { cg++; sg += fv; }
    }
    si[tid] = cg; sf[tid] = sg;
    __syncthreads();
    for (int s = 128; s > 0; s >>= 1) {
        if (tid < s) { si[tid] += si[tid + s]; sf[tid] += sf[tid + s]; }
        __syncthreads();
    }
    if (tid == 0)
        negsum[b] = sf[0] + (float)(k - si[0]) * __uint_as_float(prefix);
}

// ---------------------------------------------------------------------------
// K5: deterministic fixed-order finalize (tiny)
// ---------------------------------------------------------------------------
__global__ void mb_final(const float* __restrict__ lossl_part,
                         const float* __restrict__ posce_part,
                         const float* __restrict__ negsum,
                         const int*   __restrict__ numpos_ws,
                         float* __restrict__ out)
{
    if (threadIdx.x == 0 && blockIdx.x == 0) {
        float ll = 0.0f;
        for (int i = 0; i < B_ * K2CH; ++i) ll += lossl_part[i];
        float pc = 0.0f;
        for (int i = 0; i < B_ * K3CH; ++i) pc += posce_part[i];
        float ns = 0.0f; int np = 0;
        for (int i = 0; i < B_; ++i) { ns += negsum[i]; np += numpos_ws[i]; }
        float N = (float)np;
        out[0] = ll / N;
        out[1] = (pc + ns) / N;
    }
}

extern "C" void kernel_launch(void* const* d_in, const int* in_sizes, int n_in,
                              void* d_out, int out_size, void* d_ws, size_t ws_size,
                              hipStream_t stream) {
    const float* loc    = (const float*)d_in[0];   // [B,P,4]
    const float* conf   = (const float*)d_in[1];   // [B,P,C]
    const float* priors = (const float*)d_in[2];   // [P,4] center form
    const float* boxes  = (const float*)d_in[3];   // [B,T,4] corner form
    const int*   labels = (const int*)d_in[4];     // [B,T]
    float* out = (float*)d_out;

    char* ws = (char*)d_ws;
    int*   conf_t      = (int*)ws;    ws += (size_t)B_ * P_ * sizeof(int);
    float* mine        = (float*)ws;  ws += (size_t)B_ * P_ * sizeof(float);
    int*   bpi         = (int*)ws;    ws += (size_t)B_ * T_ * sizeof(int);
    float* lossl_part  = (float*)ws;  ws += (size_t)B_ * K2CH * sizeof(float);
    int*   poscnt_part = (int*)ws;    ws += (size_t)B_ * K2CH * sizeof(int);
    float* posce_part  = (float*)ws;  ws += (size_t)B_ * K3CH * sizeof(float);
    float* negsum      = (float*)ws;  ws += (size_t)B_ * sizeof(float);
    int*   numpos_ws   = (int*)ws;    ws += (size_t)B_ * sizeof(int);

    mb_best_prior<<<dim3(T_, B_), 256, 0, stream>>>(priors, boxes, bpi);
    mb_match<<<dim3(K2CH, B_), 256, 0, stream>>>(loc, priors, boxes, labels, bpi,
                                                 conf_t, lossl_part, poscnt_part);
    mb_ce<<<dim3(K3CH, B_), 128, 0, stream>>>(conf, conf_t, mine, posce_part);
    mb_select<<<dim3(B_), 256, 0, stream>>>(mine, poscnt_part, negsum, numpos_ws);
    mb_final<<<1, 64, 0, stream>>>(lossl_part, posce_part, negsum, numpos_ws, out);
}